// GCNLayer_32229434589582
// MI455X (gfx1250) — compile-verified
//
#include <hip/hip_runtime.h>
#include <hip/hip_bf16.h>

#define N_NODES 50000
#define D 32

typedef float v2f __attribute__((ext_vector_type(2)));
typedef float v8f __attribute__((ext_vector_type(8)));

static __device__ __forceinline__ void atomic_add_f32(float* p, float v) {
    __hip_atomic_fetch_add(p, v, __ATOMIC_RELAXED, __HIP_MEMORY_SCOPE_AGENT);
}

// ---------------------------------------------------------------- zero ws
__global__ void gcn_zero_ws(float* ws, long long count) {
    long long i = (long long)blockIdx.x * blockDim.x + threadIdx.x;
    long long stride = (long long)gridDim.x * blockDim.x;
    for (; i < count; i += stride) ws[i] = 0.0f;
}

// ------------------------------------------- stage 1: seg_sum + in-degree
// 8 threads per edge, each handles 4 consecutive features (float4 load).
__global__ void gcn_scatter_edgefeat(const float* __restrict__ ef,
                                     const int* __restrict__ dst,
                                     float* __restrict__ seg,
                                     float* __restrict__ deg,
                                     long long E) {
    long long tid = (long long)blockIdx.x * blockDim.x + threadIdx.x;
    long long e = tid >> 3;
    if (e >= E) return;
    int j = (int)(tid & 7) * 4;
    int d = dst[e];
    const float4 v = *(const float4*)(ef + e * D + j);
    float* base = seg + (long long)d * D + j;
    atomic_add_f32(base + 0, v.x);
    atomic_add_f32(base + 1, v.y);
    atomic_add_f32(base + 2, v.z);
    atomic_add_f32(base + 3, v.w);
    if (j == 0) atomic_add_f32(deg + d, 1.0f);
}

// ------------------------------------------- stage 1b: mean (in place)
__global__ void gcn_divide(float* __restrict__ seg,
                           const float* __restrict__ deg,
                           long long count) {
    long long tid = (long long)blockIdx.x * blockDim.x + threadIdx.x;
    if (tid >= count) return;
    long long node = tid >> 5;  // D == 32
    float dg = deg[node];
    seg[tid] = seg[tid] / fmaxf(dg, 1.0f);
}

// ------------------------------------------- stage 2: h = segsum(mean[src], dst)
__global__ void gcn_scatter_h(const float* __restrict__ node_mean,
                              const int* __restrict__ src,
                              const int* __restrict__ dst,
                              float* __restrict__ h,
                              long long E) {
    long long tid = (long long)blockIdx.x * blockDim.x + threadIdx.x;
    long long e = tid >> 3;
    if (e >= E) return;
    int j = (int)(tid & 7) * 4;
    int s = src[e];
    int d = dst[e];
    const float4 v = *(const float4*)(node_mean + (long long)s * D + j);
    float* base = h + (long long)d * D + j;
    atomic_add_f32(base + 0, v.x);
    atomic_add_f32(base + 1, v.y);
    atomic_add_f32(base + 2, v.z);
    atomic_add_f32(base + 3, v.w);
}

// ------------------------------------------- stage 3+4: edge_h @ W^T + b (WMMA)
// One wave handles 16 edges. out[m,n] = sum_k edge_h[m,k]*W[n,k] + b[n].
// A (16x4 f32) per ISA layout: lane L -> row m=L%16, VGPR v -> k = 2*(L/16)+v.
// B (4x16 f32) mirrors A with N across lanes: lane L -> col n=L%16, k = 2*(L/16)+v.
// C/D (16x16 f32): lane L, VGPR r -> row m = 8*(L/16)+r, col n = L%16.
__global__ __launch_bounds__(256) void gcn_edge_gemm_wmma(
    const float* __restrict__ h,     // [N, 32]
    const int* __restrict__ src,
    const int* __restrict__ dst,
    const float* __restrict__ W,     // [32, 32] row-major (D_OUT, D_IN)
    const float* __restrict__ bias,  // [32]
    float* __restrict__ out,         // [E, 32]
    long long E) {
    const int lane = threadIdx.x & 31;
    const int wave = threadIdx.x >> 5;
    const long long edgeBase = ((long long)blockIdx.x * 8 + wave) * 16;
    if (edgeBase >= E) return;  // wave-uniform; EXEC all-1s at WMMA

    const int half = lane >> 4;   // 0 or 1
    const int lid  = lane & 15;

    // B fragments from W (constant across edges): bfrag[kk][tile]
    v2f bfrag[8][2];
#pragma unroll
    for (int kk = 0; kk < 8; ++kk) {
        const int k = 4 * kk + 2 * half;
#pragma unroll
        for (int t = 0; t < 2; ++t) {
            const int n = 16 * t + lid;
            bfrag[kk][t].x = W[n * D + k];
            bfrag[kk][t].y = W[n * D + k + 1];
        }
    }

    // A fragments: edge_h = 0.5*(h[src] + h[dst]) gathered at this lane's k slots
    const long long e = edgeBase + lid;
    const float* hs = h + (long long)src[e] * D;
    const float* hd = h + (long long)dst[e] * D;
    v2f afrag[8];
#pragma unroll
    for (int kk = 0; kk < 8; ++kk) {
        const int k = 4 * kk + 2 * half;
        afrag[kk].x = 0.5f * (hs[k]     + hd[k]);
        afrag[kk].y = 0.5f * (hs[k + 1] + hd[k + 1]);
    }

    v8f c0 = {};
    v8f c1 = {};
#pragma unroll
    for (int kk = 0; kk < 8; ++kk) {
        c0 = __builtin_amdgcn_wmma_f32_16x16x4_f32(false, afrag[kk], false, bfrag[kk][0],
                                                   (short)0, c0, false, false);
        c1 = __builtin_amdgcn_wmma_f32_16x16x4_f32(false, afrag[kk], false, bfrag[kk][1],
                                                   (short)0, c1, false, false);
    }

    const float b0 = bias[lid];
    const float b1 = bias[16 + lid];
#pragma unroll
    for (int r = 0; r < 8; ++r) {
        const long long m = edgeBase + 8 * half + r;
        float* orow = out + m * D;
        orow[lid]      = c0[r] + b0;
        orow[16 + lid] = c1[r] + b1;
    }
}

// ---------------------------------------------------------------- launch
extern "C" void kernel_launch(void* const* d_in, const int* in_sizes, int n_in,
                              void* d_out, int out_size, void* d_ws, size_t ws_size,
                              hipStream_t stream) {
    const float* edge_feats = (const float*)d_in[0];
    const int*   src        = (const int*)d_in[1];
    const int*   dst        = (const int*)d_in[2];
    const float* W          = (const float*)d_in[3];
    const float* bias       = (const float*)d_in[4];
    float*       out        = (float*)d_out;

    const long long E = in_sizes[1];
    const long long N = N_NODES;

    // workspace layout: seg/node_mean [N*32] | deg [N] | h [N*32]
    float* seg = (float*)d_ws;
    float* deg = seg + N * D;
    float* h   = deg + N;
    const long long ws_count = N * (2 * D + 1);

    // 0) zero accumulators
    {
        int blocks = (int)((ws_count + 255) / 256);
        gcn_zero_ws<<<blocks, 256, 0, stream>>>((float*)d_ws, ws_count);
    }
    // 1) seg_sum + degree
    {
        long long threads = E * 8;
        int blocks = (int)((threads + 255) / 256);
        gcn_scatter_edgefeat<<<blocks, 256, 0, stream>>>(edge_feats, dst, seg, deg, E);
    }
    // 1b) mean in place
    {
        long long threads = N * D;
        int blocks = (int)((threads + 255) / 256);
        gcn_divide<<<blocks, 256, 0, stream>>>(seg, deg, threads);
    }
    // 2) h = segsum(node_mean[src], dst)
    {
        long long threads = E * 8;
        int blocks = (int)((threads + 255) / 256);
        gcn_scatter_h<<<blocks, 256, 0, stream>>>(seg, src, dst, h, E);
    }
    // 3+4) edge averaging + linear layer via WMMA (E divisible by 128)
    {
        int blocks = (int)((E + 127) / 128);
        gcn_edge_gemm_wmma<<<blocks, 256, 0, stream>>>(h, src, dst, W, bias, out, E);
    }
}